// O2_Loss_25211458027654
// MI455X (gfx1250) — compile-verified
//
#include <hip/hip_runtime.h>
#include <hip/hip_bf16.h>

typedef __attribute__((ext_vector_type(2))) float v2f;
typedef __attribute__((ext_vector_type(8))) float v8f;

// ---------------------------------------------------------------------------
// Wave32 cross-lane sum via V_WMMA_F32_16X16X4_F32.
// A (16x4 f32, 2 VGPRs/lane): lanes 0-15 -> (m=lane, k=0) in VGPR0,
// lanes 16-31 -> (m=lane-16, k=2). Partial goes in VGPR0, VGPR1 = 0, so every
// lane's value occupies a unique A cell. B = all-ones 4x16, C = 0.
// D[m][n] = sum_k A[m][k]; summing the 8 D VGPRs per lane yields rows 0-7
// (lanes 0-15) or rows 8-15 (lanes 16-31); shfl_xor(16) completes the total.
// Requires EXEC == all ones (callers guarantee full wave, uniform flow).
// ---------------------------------------------------------------------------
__device__ __forceinline__ float wave_sum_wmma(float p) {
    v2f a; a[0] = p;    a[1] = 0.0f;
    v2f b; b[0] = 1.0f; b[1] = 1.0f;
    v8f c = {};
    v8f d = __builtin_amdgcn_wmma_f32_16x16x4_f32(
        /*neg_a=*/false, a, /*neg_b=*/false, b,
        /*c_mod=*/(short)0, c, /*reuse_a=*/false, /*reuse_b=*/false);
    float s = d[0] + d[1] + d[2] + d[3] + d[4] + d[5] + d[6] + d[7];
    s += __shfl_xor(s, 16, 32);
    return s;   // total across all 32 lanes, broadcast to every lane
}

// B=16, C=5, H=W=512
#define HW_   (512u * 512u)        // 2^18
#define NPIX_ (16u * HW_)          // 2^22 = 4,194,304
#define CHW_  (5u * HW_)

__device__ __forceinline__ float pixel_penalty(const float* __restrict__ pred,
                                               const int*   __restrict__ tgt,
                                               unsigned n) {
    unsigned b  = n >> 18;            // n / HW
    unsigned hw = n & (HW_ - 1u);     // n % HW
    const float* base = pred + b * CHW_ + hw;   // all 32-bit arithmetic

    float x0 = base[0u * HW_];
    float x1 = base[1u * HW_];
    float x2 = base[2u * HW_];
    float x3 = base[3u * HW_];
    float x4 = base[4u * HW_];
    int   t  = tgt[n];

    // stabilized softmax numerators; 1/S factors out of relu(+-diff)
    float m  = fmaxf(fmaxf(fmaxf(x0, x1), fmaxf(x2, x3)), x4);
    float e0 = __expf(x0 - m);
    float e1 = __expf(x1 - m);
    float e2 = __expf(x2 - m);
    float e3 = __expf(x3 - m);
    float e4 = __expf(x4 - m);
    float invS = __builtin_amdgcn_rcpf(e0 + e1 + e2 + e3 + e4);  // v_rcp_f32

    float d1 = e1 - e0, d2 = e2 - e1, d3 = e3 - e2, d4 = e4 - e3;
    float pen;
    pen  = fmaxf((1 <= t) ? -d1 : d1, 0.0f);
    pen += fmaxf((2 <= t) ? -d2 : d2, 0.0f);
    pen += fmaxf((3 <= t) ? -d3 : d3, 0.0f);
    pen += fmaxf((4 <= t) ? -d4 : d4, 0.0f);
    return pen * invS;
}

// iters and stride are uniform; NPIX_ is exactly iters * stride, so the loop
// has a scalar trip count: no per-lane 64-bit bound checks, no exec-mask dance.
__global__ void o2_loss_main(const float* __restrict__ pred,
                             const int*   __restrict__ tgt,
                             float*       __restrict__ ws,
                             int iters, unsigned stride) {
    __shared__ float warp_sums[8];

    unsigned n = blockIdx.x * blockDim.x + threadIdx.x;
    float p = 0.0f;
    #pragma unroll 2
    for (int i = 0; i < iters; ++i) {
        p += pixel_penalty(pred, tgt, n);
        n += stride;
    }

    // per-wave reduction through the WMMA unit (full wave, uniform flow)
    float wsum = wave_sum_wmma(p);

    int lane = threadIdx.x & 31;
    int wave = threadIdx.x >> 5;
    if (lane == 0) warp_sums[wave] = wsum;
    __syncthreads();

    if (threadIdx.x == 0) {
        float s = 0.0f;
        #pragma unroll
        for (int i = 0; i < 8; ++i) s += warp_sums[i];
        ws[blockIdx.x] = s;
    }
}

__global__ void o2_loss_final(const float* __restrict__ ws, int nblocks,
                              float* __restrict__ out, float inv_npix) {
    int lane = threadIdx.x;     // launched with exactly one wave (32 threads)
    float p = 0.0f;
    for (int i = lane; i < nblocks; i += 32) p += ws[i];
    float total = wave_sum_wmma(p);
    if (lane == 0) out[0] = total * inv_npix;
}

extern "C" void kernel_launch(void* const* d_in, const int* in_sizes, int n_in,
                              void* d_out, int out_size, void* d_ws, size_t ws_size,
                              hipStream_t stream) {
    const float* pred = (const float*)d_in[0];
    const int*   tgt  = (const int*)d_in[1];
    float*       out  = (float*)d_out;
    float*       ws   = (float*)d_ws;

    // Power-of-two block count (<= 2048) so 256*nblocks divides NPIX_ = 2^22
    // exactly -> uniform per-thread trip count. 2048 blocks = 8 KB of d_ws.
    int nblocks = 2048;
    while ((size_t)nblocks * sizeof(float) > ws_size && nblocks > 1) nblocks >>= 1;

    unsigned stride = (unsigned)nblocks * 256u;
    int iters = (int)(NPIX_ / stride);      // 8 at nblocks = 2048

    o2_loss_main<<<nblocks, 256, 0, stream>>>(pred, tgt, ws, iters, stride);
    // exact: 1/(16*512*512) = 2^-22
    o2_loss_final<<<1, 32, 0, stream>>>(ws, nblocks, out, 1.0f / 4194304.0f);
}